// GraphConv_89867895702011
// MI455X (gfx1250) — compile-verified
//
#include <hip/hip_runtime.h>
#include <cstdint>

// GraphConv scatter-add:
//   out[tidx[e], :] += input[sidx[e], :] * (esgn[e] * enorm[e])
// N_NODES=100000, N_EDGES=640000, D=128 (f32).
//
// Roofline: unique working set (~115 MB) fits in the 192 MB L2; HBM-side is a
// single ~168 MB pass (~7 us @ 23.3 TB/s). Real limiter: 640K*128 = 82M f32
// L2 atomics. Strategy: one wave32 per edge (32 lanes x float4 = 512 B row),
// rows gathered via CDNA5 async global->LDS copies (double buffered,
// ASYNCcnt), scattered via non-returning global_atomic_add_f32. Per-edge
// metadata is read through the scalar (SMEM/KMcnt) path to keep the vector
// memory pipe free for atomics.

#define D_FEAT 128
#define ROW_BYTES (D_FEAT * 4)          // 512 B per row
#define WAVES_PER_BLOCK 8
#define BLOCK_THREADS (WAVES_PER_BLOCK * 32)
#define SLOT_BYTES (2 * ROW_BYTES)      // double buffer per wave

__global__ void GraphConv_zero4_kernel(float4* __restrict__ out4, int n4) {
  int i = blockIdx.x * blockDim.x + threadIdx.x;
  if (i < n4) out4[i] = make_float4(0.f, 0.f, 0.f, 0.f);
}

__global__ void GraphConv_zero1_kernel(float* __restrict__ out, int start, int n) {
  int i = start + blockIdx.x * blockDim.x + threadIdx.x;
  if (i < n) out[i] = 0.f;
}

// Issue an async copy of this lane's 16B chunk of row `sidx` into LDS.
__device__ __forceinline__ void async_row_chunk_to_lds(
    const float* __restrict__ input, int sidx, uint32_t lds_addr, int lane) {
  unsigned long long gaddr =
      (unsigned long long)(uintptr_t)(input + (long long)sidx * D_FEAT) +
      (unsigned)(lane * 16);
  // VDST = LDS byte address (VGPR), VADDR = 64-bit global address (VGPR pair)
  asm volatile("global_load_async_to_lds_b128 %0, %1, off"
               :: "v"(lds_addr), "v"(gaddr)
               : "memory");
}

__global__ void __launch_bounds__(BLOCK_THREADS)
GraphConv_scatter_kernel(const float* __restrict__ input,
                         const int* __restrict__ eidx,   // [2, E] row-major
                         const float* __restrict__ enorm,
                         const float* __restrict__ esgn,
                         float* __restrict__ out,
                         long long n_edges) {
  __shared__ char smem[WAVES_PER_BLOCK * SLOT_BYTES];

  const int lane = threadIdx.x & 31;
  // Force wave-uniformity so per-edge metadata loads become scalar (SMEM).
  const int wib  = __builtin_amdgcn_readfirstlane((int)(threadIdx.x >> 5));
  const long long wid = (long long)blockIdx.x * WAVES_PER_BLOCK + wib;
  const long long nw  = (long long)gridDim.x * WAVES_PER_BLOCK;

  char* slot = smem + wib * SLOT_BYTES;
  // Low 32 bits of a flat pointer into LDS == LDS byte offset (aperture in MSBs).
  const uint32_t lds_lane_base =
      (uint32_t)(uintptr_t)slot + (uint32_t)(lane * 16);

  long long e = wid;
  if (e >= n_edges) return;

  // Prologue: fetch first edge's row into buffer 0.
  async_row_chunk_to_lds(input, eidx[e], lds_lane_base, lane);

  int buf = 0;
  for (; e < n_edges; e += nw) {
    const long long en = e + nw;
    if (en < n_edges) {
      // Prefetch next edge's row into the other buffer, then wait for current.
      async_row_chunk_to_lds(input, eidx[en],
                             lds_lane_base + (uint32_t)((buf ^ 1) * ROW_BYTES),
                             lane);
      asm volatile("s_wait_asynccnt 1" ::: "memory");
    } else {
      asm volatile("s_wait_asynccnt 0" ::: "memory");
    }

    const int   tidx = eidx[n_edges + e];
    const float w    = esgn[e] * enorm[e];

    const float4 v = *reinterpret_cast<const float4*>(
        slot + buf * ROW_BYTES + lane * 16);

    float* dst = out + (long long)tidx * D_FEAT + lane * 4;
    __hip_atomic_fetch_add(dst + 0, v.x * w, __ATOMIC_RELAXED, __HIP_MEMORY_SCOPE_AGENT);
    __hip_atomic_fetch_add(dst + 1, v.y * w, __ATOMIC_RELAXED, __HIP_MEMORY_SCOPE_AGENT);
    __hip_atomic_fetch_add(dst + 2, v.z * w, __ATOMIC_RELAXED, __HIP_MEMORY_SCOPE_AGENT);
    __hip_atomic_fetch_add(dst + 3, v.w * w, __ATOMIC_RELAXED, __HIP_MEMORY_SCOPE_AGENT);

    buf ^= 1;
  }
}

extern "C" void kernel_launch(void* const* d_in, const int* in_sizes, int n_in,
                              void* d_out, int out_size, void* d_ws, size_t ws_size,
                              hipStream_t stream) {
  (void)n_in; (void)d_ws; (void)ws_size;

  const float* input = (const float*)d_in[0];
  const int*   eidx  = (const int*)d_in[1];   // [2, E] int indices
  const float* enorm = (const float*)d_in[2];
  const float* esgn  = (const float*)d_in[3];
  float* out = (float*)d_out;

  const long long n_edges = (long long)in_sizes[1] / 2;

  // 1) Zero the output (harness poisons d_out). Vectorized body + rare tail.
  {
    int n4 = out_size / 4;
    if (n4 > 0) {
      int blocks = (n4 + BLOCK_THREADS - 1) / BLOCK_THREADS;
      GraphConv_zero4_kernel<<<blocks, BLOCK_THREADS, 0, stream>>>(
          (float4*)out, n4);
    }
    int tail = out_size - n4 * 4;
    if (tail > 0) {
      GraphConv_zero1_kernel<<<1, 32, 0, stream>>>(out, n4 * 4, out_size);
    }
  }

  // 2) Edge scatter: 64K waves -> ~10 edges each (pipeline depth + occupancy).
  {
    int blocks = 8192;
    GraphConv_scatter_kernel<<<blocks, BLOCK_THREADS, 0, stream>>>(
        input, eidx, enorm, esgn, out, n_edges);
  }
}